// GCN_36464272343074
// MI455X (gfx1250) — compile-verified
//
#include <hip/hip_runtime.h>
#include <hip/hip_bf16.h>
#include <math.h>

typedef float v2f __attribute__((ext_vector_type(2)));
typedef float v8f __attribute__((ext_vector_type(8)));

#define NNODES 100000
#define CIN    128

// ---------------------------------------------------------------------------
// Degree / normalization
// ---------------------------------------------------------------------------
__global__ void k_fill1(float* __restrict__ deg, int n) {
    int i = blockIdx.x * blockDim.x + threadIdx.x;
    if (i < n) deg[i] = 1.0f;                       // self-loop contributes 1
}

__global__ void k_degacc(const int* __restrict__ ei, float* __restrict__ deg, int E) {
    int e = blockIdx.x * blockDim.x + threadIdx.x;
    if (e < E) atomicAdd(&deg[ei[E + e]], 1.0f);    // dst = ei[1][e]
}

__global__ void k_rsqrt(float* __restrict__ d, int n) {
    int i = blockIdx.x * blockDim.x + threadIdx.x;
    if (i < n) d[i] = rsqrtf(d[i]);                 // deg >= 1 always
}

// ---------------------------------------------------------------------------
// Dense GEMM via V_WMMA_F32_16X16X4_F32.
// Block = 128 threads = 4 waves. Each wave owns one 16-col tile and TWO
// 16-row M-tiles (32 rows), so every B fragment feeds two WMMAs.
// K and NOUT are compile-time constants: the k-loop fully unrolls and all
// A/B loads become immediate-offset global_load's from fixed base pointers
// (no per-step 64-bit address math).
// Epilogue: S[row] = dinv[row] * (X@W)[row]  (pre-scaled message)
//           G[row] = S[row]                  (self-loop init of accumulator)
// ---------------------------------------------------------------------------
template <int K, int NOUT>
__global__ __launch_bounds__(128) void k_gemm_wmma(
    const float* __restrict__ X, const float* __restrict__ W,
    const float* __restrict__ dinv,
    float* __restrict__ S, float* __restrict__ G)
{
    const int lane   = threadIdx.x & 31;
    const int waveId = threadIdx.x >> 5;
    const int half   = lane >> 4;        // 0: K+0/K+1 ; 1: K+2/K+3
    const int l15    = lane & 15;

    const long m0   = (long)blockIdx.x * 32;                 // first row of tile 0
    const int  colB = (blockIdx.y * 4 + waveId) * 16 + l15;  // B/D: lane = N
    const float* __restrict__ xrow0 = X + (m0 + l15) * K + half * 2;   // A: lane = M
    const float* __restrict__ xrow1 = xrow0 + 16 * K;
    const float* __restrict__ wcol  = W + (size_t)half * 2 * NOUT + colB;

    v8f acc0 = {};
    v8f acc1 = {};
#pragma unroll
    for (int k0 = 0; k0 < K; k0 += 4) {
        v2f b;
        b.x = wcol[(size_t)k0 * NOUT];                 // immediate offsets
        b.y = wcol[(size_t)k0 * NOUT + NOUT];
        v2f a0, a1;
        a0.x = xrow0[k0];
        a0.y = xrow0[k0 + 1];
        a1.x = xrow1[k0];
        a1.y = xrow1[k0 + 1];
        acc0 = __builtin_amdgcn_wmma_f32_16x16x4_f32(
            false, a0, false, b, (short)0, acc0, false, false);
        acc1 = __builtin_amdgcn_wmma_f32_16x16x4_f32(
            false, a1, false, b, (short)0, acc1, false, false);
    }

    // C/D layout: VGPR r, lanes 0-15 -> M=r ; lanes 16-31 -> M=r+8
    const long  rbase = m0 + half * 8;
    float* __restrict__ sp = S + (size_t)rbase * NOUT + colB;
    float* __restrict__ gp = G + (size_t)rbase * NOUT + colB;
#pragma unroll
    for (int r = 0; r < 8; ++r) {
        float v = dinv[rbase + r] * acc0[r];
        sp[(size_t)r * NOUT] = v;
        gp[(size_t)r * NOUT] = v;

        v = dinv[rbase + 16 + r] * acc1[r];
        sp[(size_t)(r + 16) * NOUT] = v;
        gp[(size_t)(r + 16) * NOUT] = v;
    }
}

// ---------------------------------------------------------------------------
// Edge scatter: one wave32 per edge, VPL floats per lane (C = 32*VPL).
// G[dst] += S[src]  via global_atomic_add_f32. S and G together fit in the
// 192MB L2, so these atomics resolve at L2 rate.
// ---------------------------------------------------------------------------
template <int VPL>
__global__ __launch_bounds__(256) void k_scatter(
    const int* __restrict__ ei, const float* __restrict__ S,
    float* __restrict__ G, int E)
{
    const int w    = (blockIdx.x * 256 + threadIdx.x) >> 5;
    const int lane = threadIdx.x & 31;
    if (w >= E) return;
    const int s = ei[w];       // src
    const int d = ei[E + w];   // dst
    const int C = VPL * 32;
    const float* __restrict__ sp = S + (size_t)s * C + lane * VPL;
    float* __restrict__       gp = G + (size_t)d * C + lane * VPL;
    float v[VPL];
#pragma unroll
    for (int i = 0; i < VPL; ++i) v[i] = sp[i];   // contiguous -> b64/b128 load
#pragma unroll
    for (int i = 0; i < VPL; ++i) atomicAdd(gp + i, v[i]);
}

// ---------------------------------------------------------------------------
// Post-scale + bias + ELU, in place:  G[i] = elu(dinv[row]*G[i] + b[col])
// ---------------------------------------------------------------------------
__global__ void k_bias_elu(float* __restrict__ G, const float* __restrict__ dinv,
                           const float* __restrict__ b, int total, int logC)
{
    int i = blockIdx.x * blockDim.x + threadIdx.x;
    if (i >= total) return;
    const int row = i >> logC;
    const int col = i & ((1 << logC) - 1);
    float v = dinv[row] * G[i] + b[col];
    G[i] = v > 0.0f ? v : (__expf(v) - 1.0f);
}

// ---------------------------------------------------------------------------
// Layer 3: 64 -> 1 matvec with dinv pre-scale; G3 initialized to S3.
// ---------------------------------------------------------------------------
__global__ void k_matvec64(const float* __restrict__ H, const float* __restrict__ W3,
                           const float* __restrict__ dinv,
                           float* __restrict__ S3, float* __restrict__ G3, int n)
{
    int i = blockIdx.x * blockDim.x + threadIdx.x;
    if (i >= n) return;
    const float* __restrict__ h = H + (size_t)i * 64;
    float acc = 0.0f;
#pragma unroll
    for (int c = 0; c < 64; ++c) acc = fmaf(h[c], W3[c], acc);
    const float v = dinv[i] * acc;
    S3[i] = v;
    G3[i] = v;
}

__global__ void k_scatter1f(const int* __restrict__ ei, const float* __restrict__ S3,
                            float* __restrict__ G3, int E)
{
    int e = blockIdx.x * blockDim.x + threadIdx.x;
    if (e < E) atomicAdd(&G3[ei[E + e]], S3[ei[e]]);
}

__global__ void k_final(const float* __restrict__ G3, const float* __restrict__ dinv,
                        const float* __restrict__ b3, float* __restrict__ out, int n)
{
    int i = blockIdx.x * blockDim.x + threadIdx.x;
    if (i < n) out[i] = dinv[i] * G3[i] + b3[0];
}

// ---------------------------------------------------------------------------
extern "C" void kernel_launch(void* const* d_in, const int* in_sizes, int n_in,
                              void* d_out, int out_size, void* d_ws, size_t ws_size,
                              hipStream_t stream)
{
    const float* x  = (const float*)d_in[0];
    const int*   ei = (const int*)  d_in[1];   // [2,E]: src then dst (int32)
    const float* W1 = (const float*)d_in[2];
    const float* b1 = (const float*)d_in[3];
    const float* W2 = (const float*)d_in[4];
    const float* b2 = (const float*)d_in[5];
    const float* W3 = (const float*)d_in[6];
    const float* b3 = (const float*)d_in[7];
    float* out = (float*)d_out;

    const int N = NNODES;
    const int C = CIN;
    const int E = in_sizes[1] / 2;

    // Workspace layout (floats): dinv | s3 | g3 | P (N*128) | Q (N*128)
    float* ws   = (float*)d_ws;
    float* dinv = ws;
    float* s3   = ws + (size_t)N;
    float* g3   = ws + (size_t)2 * N;
    float* P    = ws + (size_t)3 * N;
    float* Q    = P + (size_t)N * C;

    const int TB = 256;

    // --- normalization: deg (with self-loop) -> dinv = deg^-1/2 ------------
    k_fill1 <<<(N + TB - 1) / TB, TB, 0, stream>>>(dinv, N);
    k_degacc<<<(E + TB - 1) / TB, TB, 0, stream>>>(ei, dinv, E);
    k_rsqrt <<<(N + TB - 1) / TB, TB, 0, stream>>>(dinv, N);

    // --- layer 1: C=128 -> 128 --------------------------------------------
    {
        dim3 grid(N / 32, C / 64);              // 3125 x 2; 4 waves x 2 M-tiles
        k_gemm_wmma<128, 128><<<grid, 128, 0, stream>>>(x, W1, dinv, P, Q);
        k_scatter<4><<<(E * 8 + TB - 1) / TB, TB, 0, stream>>>(ei, P, Q, E);
        k_bias_elu<<<(N * C + TB - 1) / TB, TB, 0, stream>>>(Q, dinv, b1, N * C, 7);
    }

    // --- layer 2: C=128 -> 64 ---------------------------------------------
    {
        float* S2 = P;
        float* G2 = P + (size_t)N * 64;
        dim3 grid(N / 32, 1);                   // 3125 x 1
        k_gemm_wmma<128, 64><<<grid, 128, 0, stream>>>(Q, W2, dinv, S2, G2);
        k_scatter<2><<<(E * 8 + TB - 1) / TB, TB, 0, stream>>>(ei, S2, G2, E);
        k_bias_elu<<<(N * 64 + TB - 1) / TB, TB, 0, stream>>>(G2, dinv, b2, N * 64, 6);
    }

    // --- layer 3: 64 -> 1 --------------------------------------------------
    {
        const float* H2 = P + (size_t)N * 64;
        k_matvec64<<<(N + TB - 1) / TB, TB, 0, stream>>>(H2, W3, dinv, s3, g3, N);
        k_scatter1f<<<(E + TB - 1) / TB, TB, 0, stream>>>(ei, s3, g3, E);
        k_final<<<(N + TB - 1) / TB, TB, 0, stream>>>(g3, dinv, b3, out, N);
    }
}